// GaussianScene_11476152615027
// MI455X (gfx1250) — compile-verified
//
#include <hip/hip_runtime.h>
#include <cstdint>

#define TILE 256

typedef uint32_t u32x4 __attribute__((ext_vector_type(4)));
typedef uint32_t u32x8 __attribute__((ext_vector_type(8)));

// ---- CDNA5 async global->LDS dword copies (ASYNCcnt) for the ragged tail ----
__device__ __forceinline__ void async_copy_b32(uint32_t lds_byte_addr, uint64_t gaddr) {
    asm volatile("global_load_async_to_lds_b32 %0, %1, off"
                 :: "v"(lds_byte_addr), "v"(gaddr) : "memory");
}
__device__ __forceinline__ void wait_async0() {
    asm volatile("s_wait_asynccnt 0" ::: "memory");
}
// Generic pointer to a __shared__ object: low 32 bits = wave-relative LDS byte
// address (flat-aperture truncation rule, CDNA5 ISA).
__device__ __forceinline__ uint32_t lds_addr_of(const void* p) {
    return (uint32_t)(uintptr_t)p;
}

// ---- CDNA5 Tensor Data Mover: contiguous 1-D tile, global -> LDS ----
// D# group0: [1:0]=count=1, [63:32]=lds_addr, [120:64]=global_addr, [127:126]=type=2
// D# group1: [17:16]=data_size(2 -> 4B), [79:48]=tensor_dim0, [95:80]=tensor_dim1=1,
//            [127:112]=tile_dim0, [143:128]=tile_dim1=1, [207:160]=tensor_dim0_stride
__device__ __forceinline__ void tdm_load_1d(uint32_t lds_byte_addr, uint64_t gaddr,
                                            uint32_t nelem /* f32 elements, <= 65535 */) {
    const u32x4 g0 = {
        1u,                                                        // count=1, user D#
        lds_byte_addr,                                             // lds_addr
        (uint32_t)gaddr,                                           // global_addr[31:0]
        (((uint32_t)(gaddr >> 32)) & 0x01FFFFFFu) | 0x80000000u    // ga[56:32] | type=2
    };
    const u32x8 g1 = {
        2u << 16,                                   // data_size = 4 bytes
        (nelem & 0xFFFFu) << 16,                    // tensor_dim0[15:0]  -> bits 63:48
        ((nelem >> 16) & 0xFFFFu) | (1u << 16),     // tensor_dim0[31:16], tensor_dim1=1
        (nelem & 0xFFFFu) << 16,                    // tile_dim0          -> bits 127:112
        1u,                                         // tile_dim1 = 1, tile_dim2 = 0
        nelem,                                      // tensor_dim0_stride[31:0]
        0u,                                         // stride0[47:32]=0, stride1[15:0]=0
        0u
    };
    // Groups 2/3 omitted (NULL): tensors up to 2-D. Portable inline-asm form.
    asm volatile("tensor_load_to_lds %0, %1" :: "s"(g0), "s"(g1) : "memory");
}

__global__ __launch_bounds__(TILE)
void splat_project_fwd(const float* __restrict__ pts,
                       const float* __restrict__ quat,
                       const float* __restrict__ scl,
                       const float* __restrict__ Em,   // 4x4 row-major
                       const float* __restrict__ Km,   // 3x4 row-major
                       const int*   __restrict__ hgt,
                       const int*   __restrict__ wid,
                       float* __restrict__ out, int N)
{
    __shared__ float s_pts[TILE * 3];
    __shared__ float s_quat[TILE * 4];
    __shared__ float s_scl[TILE * 3];

    const int tid  = threadIdx.x;
    const int base = blockIdx.x * TILE;
    if (base >= N) return;
    const int rem = min(TILE, N - base);

    const uint64_t gp = (uint64_t)(uintptr_t)(pts  + (size_t)base * 3);
    const uint64_t gq = (uint64_t)(uintptr_t)(quat + (size_t)base * 4);
    const uint64_t gs = (uint64_t)(uintptr_t)(scl  + (size_t)base * 3);

    if (rem == TILE) {
        // Full tile: one wave issues three TDM descriptors (10 KB total staging).
        if (tid < 32) {
            tdm_load_1d(lds_addr_of(s_quat), gq, TILE * 4);
            tdm_load_1d(lds_addr_of(s_pts),  gp, TILE * 3);
            tdm_load_1d(lds_addr_of(s_scl),  gs, TILE * 3);
            __builtin_amdgcn_s_wait_tensorcnt(0);   // issuing wave waits for DMA
        }
    } else {
        // Ragged tail: byte-exact dword async copies by all lanes.
        for (int j = tid; j < rem * 3; j += TILE) {
            async_copy_b32(lds_addr_of(s_pts) + j * 4, gp + (uint64_t)j * 4);
            async_copy_b32(lds_addr_of(s_scl) + j * 4, gs + (uint64_t)j * 4);
        }
        for (int j = tid; j < rem * 4; j += TILE) {
            async_copy_b32(lds_addr_of(s_quat) + j * 4, gq + (uint64_t)j * 4);
        }
        wait_async0();
    }
    __syncthreads();     // staged tile visible to all waves

    if (tid >= rem) return;
    const int i = base + tid;

    // ---- uniform camera constants (scalar loads) ----
    const float e00 = Em[0],  e01 = Em[1],  e02 = Em[2],  e03 = Em[3];
    const float e10 = Em[4],  e11 = Em[5],  e12 = Em[6],  e13 = Em[7];
    const float e20 = Em[8],  e21 = Em[9],  e22 = Em[10], e23 = Em[11];
    const float k00 = Km[0],  k01 = Km[1],  k02 = Km[2],  k03 = Km[3];
    const float k10 = Km[4],  k11 = Km[5],  k12 = Km[6],  k13 = Km[7];
    const float k20 = Km[8],  k21 = Km[9],  k22 = Km[10], k23 = Km[11];
    const float fx = k00, fy = k11;
    const float Wf = (float)wid[0], Hf = (float)hgt[0];
    const float R2D = 57.29577951308232f;
    const float fovx = 2.0f * atanf(Wf / (2.0f * fx)) * R2D;
    const float fovy = 2.0f * atanf(Hf / (2.0f * fy)) * R2D;
    const float lx = 1.3f * fovx, ly = 1.3f * fovy;

    // ---- per-point data from LDS ----
    const float qw = s_quat[tid * 4 + 0], qx = s_quat[tid * 4 + 1];
    const float qy = s_quat[tid * 4 + 2], qz = s_quat[tid * 4 + 3];
    const float px = s_pts[tid * 3 + 0],  py = s_pts[tid * 3 + 1],  pz = s_pts[tid * 3 + 2];
    const float sx = s_scl[tid * 3 + 0],  sy = s_scl[tid * 3 + 1],  sz = s_scl[tid * 3 + 2];

    // quaternion -> rotation
    const float qn = 1.0f / sqrtf(qw*qw + qx*qx + qy*qy + qz*qz);
    const float w = qw*qn, x = qx*qn, y = qy*qn, z = qz*qn;
    const float xx = x*x, yy = y*y, zz = z*z;
    const float xy = x*y, xz = x*z, yz = y*z;
    const float wx = w*x, wy = w*y, wz = w*z;
    const float r00 = 1.0f - 2.0f*(yy+zz), r01 = 2.0f*(xy-wz),        r02 = 2.0f*(xz+wy);
    const float r10 = 2.0f*(xy+wz),        r11 = 1.0f - 2.0f*(xx+zz), r12 = 2.0f*(yz-wx);
    const float r20 = 2.0f*(xz-wy),        r21 = 2.0f*(yz+wx),        r22 = 1.0f - 2.0f*(xx+yy);

    // M = R * diag(s); V = M M^T (symmetric 3x3)
    const float m00 = r00*sx, m01 = r01*sy, m02 = r02*sz;
    const float m10 = r10*sx, m11 = r11*sy, m12 = r12*sz;
    const float m20 = r20*sx, m21 = r21*sy, m22 = r22*sz;
    const float v00 = m00*m00 + m01*m01 + m02*m02;
    const float v01 = m00*m10 + m01*m11 + m02*m12;
    const float v02 = m00*m20 + m01*m21 + m02*m22;
    const float v11 = m10*m10 + m11*m11 + m12*m12;
    const float v12 = m10*m20 + m11*m21 + m12*m22;
    const float v22 = m20*m20 + m21*m21 + m22*m22;

    // camera-space point, projection
    const float c0 = e00*px + e01*py + e02*pz + e03;
    const float c1 = e10*px + e11*py + e12*pz + e13;
    const float c2 = e20*px + e21*py + e22*pz + e23;
    const float ph0 = k00*c0 + k01*c1 + k02*c2 + k03;
    const float ph1 = k10*c0 + k11*c1 + k12*c2 + k13;
    const float ph2 = k20*c0 + k21*c1 + k22*c2 + k23;
    const float ppx = ph0 / ph2, ppy = ph1 / ph2;

    // Jacobian rows with clipped tan-angles (degree bounds, per reference)
    const float tx = fminf(fmaxf(c0 / c2, -lx), lx);
    const float ty = fminf(fmaxf(c1 / c2, -ly), ly);
    // T = J @ E[:3,:3]^T
    const float t00 = fx*e00 - fx*tx*e02;
    const float t01 = fx*e10 - fx*tx*e12;
    const float t02 = fx*e20 - fx*tx*e22;
    const float t10 = fy*e01 - fy*ty*e02;
    const float t11 = fy*e11 - fy*ty*e12;
    const float t12 = fy*e21 - fy*ty*e22;

    // cov2d = T V T^T
    const float u00 = t00*v00 + t01*v01 + t02*v02;
    const float u01 = t00*v01 + t01*v11 + t02*v12;
    const float u02 = t00*v02 + t01*v12 + t02*v22;
    const float u10 = t10*v00 + t11*v01 + t12*v02;
    const float u11 = t10*v01 + t11*v11 + t12*v12;
    const float u12 = t10*v02 + t11*v12 + t12*v22;
    const float cA = u00*t00 + u01*t01 + u02*t02;
    const float cB = u00*t10 + u01*t11 + u02*t12;
    const float cC = u10*t00 + u11*t01 + u12*t02;
    const float cD = u10*t10 + u11*t11 + u12*t12;

    // outputs: [0,2N) projected_points, [2N,6N) projected_covariance
    ((float2*)out)[i] = make_float2(ppx, ppy);
    ((float4*)(out + (size_t)2 * N))[i] = make_float4(cA, cB, cC, cD);
}

extern "C" void kernel_launch(void* const* d_in, const int* in_sizes, int n_in,
                              void* d_out, int out_size, void* d_ws, size_t ws_size,
                              hipStream_t stream) {
    const float* pts  = (const float*)d_in[0];
    // d_in[1] = colors, d_in[2] = opacity : unused by reference forward
    const float* quat = (const float*)d_in[3];
    const float* scl  = (const float*)d_in[4];
    const float* Em   = (const float*)d_in[5];
    const float* Km   = (const float*)d_in[6];
    const int*   hgt  = (const int*)d_in[7];
    const int*   wid  = (const int*)d_in[8];
    const int N = in_sizes[0] / 3;
    const int grid = (N + TILE - 1) / TILE;
    hipLaunchKernelGGL(splat_project_fwd, dim3(grid), dim3(TILE), 0, stream,
                       pts, quat, scl, Em, Km, hgt, wid, (float*)d_out, N);
}